// LogRatio_89936615178820
// MI455X (gfx1250) — compile-verified
//
#include <hip/hip_runtime.h>
#include <hip/hip_bf16.h>

typedef float v2f __attribute__((ext_vector_type(2)));
typedef float v8f __attribute__((ext_vector_type(8)));

#define N_PTS   2048
#define D_EMB   128
#define KK      4
#define N_TILES (N_PTS / 16)
#define J_SLICES 8

// c_m = 0.1*(ln(0.1+1e-6) - ln(0.1^(5-m)+1e-6)), m=0..3 (double-derived constants)
#define C0 0.9115040192f
#define C1 0.6897814948f
#define C2 0.4604180686f
#define C3 0.2302495098f

__global__ void lr_zero_out(float* out, int n) {
    int t = blockIdx.x * blockDim.x + threadIdx.x;
    if (t < n) out[t] = 0.0f;
}

__global__ __launch_bounds__(32) void lr_tile_kernel(const float* __restrict__ X,
                                                     const int*   __restrict__ labels,
                                                     float*       __restrict__ out) {
    const int lane = threadIdx.x;      // 0..31, one full wave32 (EXEC all ones for WMMA)
    const int g    = lane >> 4;        // lane half
    const int n16  = lane & 15;        // M (for A) and N (for B/C) sub-index
    const int i0   = blockIdx.x * 16;  // row tile

    // targets for the 8 C-rows this lane owns: row = r + 8*g
    int ti[8];
#pragma unroll
    for (int r = 0; r < 8; ++r)
        ti[r] = labels[(size_t)(i0 + r + 8 * g) * KK + 0];

    // ---- preload ALL A fragments once: invariant across the j-tile loop ----
    // per-lane element: A[M=n16][K = 4u + v + 2g] -> float2 at arow + 4u
    const float* arow = X + (size_t)(i0 + n16) * D_EMB + 2 * g;
    v2f afrag[32];
#pragma unroll
    for (int u = 0; u < 32; ++u)
        afrag[u] = *(const v2f*)(arow + 4 * u);

    // 9 per-row accumulators (partial over this lane's N column, this wave's j-tiles)
    float pn[8]  = {}, s1[8]  = {}, s2[8] = {};
    float nn[8]  = {}, cnn[8] = {}, c2nn[8] = {};
    float t1[8]  = {}, ct1[8] = {}, t2[8] = {};

    for (int jt = (int)blockIdx.y; jt < N_TILES; jt += J_SLICES) {
        const int j0 = jt * 16;
        const float* brow = X + (size_t)(j0 + n16) * D_EMB + 2 * g;

        // prefetch this wave's next j-tile rows into cache
        if (jt + J_SLICES < N_TILES)
            __builtin_prefetch(X + (size_t)(j0 + J_SLICES * 16 + n16) * D_EMB, 0, 0);

        // labels of column j = j0 + n16 (same for both lane halves); issue early
        const int* lj = labels + (size_t)(j0 + n16) * KK;
        const int tj0 = lj[0], tj1 = lj[1], tj2 = lj[2], tj3 = lj[3];
        const int jidx = j0 + n16;

        // ---- stage all 32 B fragments in distinct regs: loads batch, waits stage ----
        v2f bfrag[32];
#pragma unroll
        for (int u = 0; u < 32; ++u)
            bfrag[u] = *(const v2f*)(brow + 4 * u);

        // ---- 16x16 Gram tile via fp32 WMMA: 32 steps of K=4 ----
        v8f acc = {0.f, 0.f, 0.f, 0.f, 0.f, 0.f, 0.f, 0.f};
#pragma unroll
        for (int u = 0; u < 32; ++u)
            acc = __builtin_amdgcn_wmma_f32_16x16x4_f32(
                /*neg_a=*/false, afrag[u], /*neg_b=*/false, bfrag[u],
                /*c_mod=*/(short)0, acc, /*reuse_a=*/false, /*reuse_b=*/false);

        // ---- branchless classify + accumulate; C layout: vgpr r -> row r+8g ----
#pragma unroll
        for (int r = 0; r < 8; ++r) {
            const float a  = __logf(acc[r] + 1e-6f);
            const float a2 = a * a;
            const int   t  = ti[r];
            const bool P0 = (tj0 == t);
            const bool P1 = (tj1 == t);
            const bool P2 = (tj2 == t);
            const bool P3 = (tj3 == t);

            // positive: same target, off-diagonal (select -> v_cndmask, no branches)
            const float posw = (P0 && ((i0 + r + 8 * g) != jidx)) ? 1.0f : 0.0f;
            pn[r] += posw;
            s1[r]  = fmaf(posw, a,  s1[r]);
            s2[r]  = fmaf(posw, a2, s2[r]);

            // mutually-exclusive negative categories with folded c_m
            const float cm   = !P3 ? C0 : (!P2 ? C1 : (!P1 ? C2 : C3));
            const float negw = (P0 && P1 && P2 && P3) ? 0.0f : 1.0f;
            const float wcm  = negw * cm;
            nn[r]  += negw;
            cnn[r] += wcm;
            c2nn[r] = fmaf(wcm, cm, c2nn[r]);
            t1[r]   = fmaf(negw, a,  t1[r]);
            ct1[r]  = fmaf(wcm,  a,  ct1[r]);
            t2[r]   = fmaf(negw, a2, t2[r]);
        }
    }

    // ---- reduce each row's 9 sums over its 16-lane half, combine, atomic-add ----
    float contribHalf = 0.0f;
#pragma unroll
    for (int r = 0; r < 8; ++r) {
        float vpn = pn[r], vs1 = s1[r], vs2 = s2[r];
        float vnn = nn[r], vcnn = cnn[r], vc2 = c2nn[r];
        float vt1 = t1[r], vct1 = ct1[r], vt2 = t2[r];
#pragma unroll
        for (int off = 1; off < 16; off <<= 1) {
            vpn  += __shfl_xor(vpn,  off, 32);
            vs1  += __shfl_xor(vs1,  off, 32);
            vs2  += __shfl_xor(vs2,  off, 32);
            vnn  += __shfl_xor(vnn,  off, 32);
            vcnn += __shfl_xor(vcnn, off, 32);
            vc2  += __shfl_xor(vc2,  off, 32);
            vt1  += __shfl_xor(vt1,  off, 32);
            vct1 += __shfl_xor(vct1, off, 32);
            vt2  += __shfl_xor(vt2,  off, 32);
        }
        // replicated across the half; every lane computes the same value
        contribHalf += vs2 * vnn - 2.0f * vs1 * vcnn + vpn * vc2
                     - 2.0f * vs1 * vt1 + 2.0f * vpn * vct1 + vpn * vt2;
    }
    const float tot = contribHalf + __shfl_xor(contribHalf, 16, 32);
    if (lane == 0) atomicAdd(out, tot);
}

extern "C" void kernel_launch(void* const* d_in, const int* in_sizes, int n_in,
                              void* d_out, int out_size, void* d_ws, size_t ws_size,
                              hipStream_t stream) {
    const float* X      = (const float*)d_in[0];
    const int*   labels = (const int*)d_in[1];
    float*       out    = (float*)d_out;

    lr_zero_out<<<1, 32, 0, stream>>>(out, out_size);

    dim3 grid(N_TILES, J_SLICES);  // 128 row tiles x 8 j-slices, one wave32 each
    lr_tile_kernel<<<grid, 32, 0, stream>>>(X, labels, out);
}